// RNN_71107478553197
// MI455X (gfx1250) — compile-verified
//
#include <hip/hip_runtime.h>

// Elman RNN for MI455X (gfx1250, wave32, WMMA + TDM), fp32 throughout.
// Phase 1: xproj = x @ W_x^T + b_x           (parallel GEMM, whole chip)
// Phase 2: fused scan: h_t = relu(xp_t + h_{t-1} @ W_h^T)
//          with y_{t-1} = h_{t-1} @ W_y^T + b_y pipelined into the same loop,
//          and the next step's xp tile DMA'd into LDS by the Tensor Data
//          Mover (TENSORcnt) while the WMMA chains run.

typedef __attribute__((ext_vector_type(2))) float        v2f;
typedef __attribute__((ext_vector_type(4))) float        v4f;
typedef __attribute__((ext_vector_type(8))) float        v8f;
typedef __attribute__((ext_vector_type(4))) unsigned int v4u;
typedef __attribute__((ext_vector_type(4))) int          v4i;
typedef __attribute__((ext_vector_type(8))) int          v8i;

#define RNN_T 512
#define RNN_N 256
#define RNN_C 128
#define RNN_D 128
#define RNN_K 128

#define WMMA4(ACC, A, B) \
    __builtin_amdgcn_wmma_f32_16x16x4_f32(false, (A), false, (B), (short)0, (ACC), false, false)

// ---------------------------------------------------------------------------
// out[M x 128] = A[M x 128] @ W[128 x 128]^T + bias[128]
// 256 threads = 8 waves; wave = 16 rows x 128 cols; W staged in LDS.
// ---------------------------------------------------------------------------
__global__ __launch_bounds__(256) void gemm_xWT_bias(
    const float* __restrict__ A,
    const float* __restrict__ W,
    const float* __restrict__ bias,
    float* __restrict__ out)
{
    extern __shared__ float smem[];     // 64 KB: W
    float* wlds = smem;

    const int tid = threadIdx.x;
    {
        const v4f* src = (const v4f*)W;
        v4f*       dst = (v4f*)wlds;
#pragma unroll
        for (int i = 0; i < 16; ++i) dst[tid + i * 256] = src[tid + i * 256];
    }
    __syncthreads();

    const int lane  = tid & 31;
    const int wave  = tid >> 5;
    const int row16 = lane & 15;
    const int khalf = (lane >> 4) << 1;   // 0 or 2
    const int hi8   = (lane >> 4) << 3;   // 0 or 8
    const long m0   = (long)blockIdx.x * 128 + wave * 16;

    v2f afrag[32];
    const float* arow = A + (m0 + row16) * 128 + khalf;
#pragma unroll
    for (int ks = 0; ks < 32; ++ks)
        afrag[ks] = *(const v2f*)(arow + ks * 4);

#pragma unroll
    for (int j = 0; j < 8; ++j) {
        v8f acc = {};
        const float* brow = wlds + (j * 16 + row16) * 128 + khalf;
#pragma unroll
        for (int ks = 0; ks < 32; ++ks) {
            v2f bfrag = *(const v2f*)(brow + ks * 4);
            acc = WMMA4(acc, afrag[ks], bfrag);
        }
        const float bv = bias[j * 16 + row16];
        float* orow = out + (m0 + hi8) * 128 + j * 16 + row16;
#pragma unroll
        for (int r = 0; r < 8; ++r)
            orow[r * 128] = acc[r] + bv;
    }
}

// ---------------------------------------------------------------------------
// Fused recurrent scan + output projection, with TDM prefetch of xp tiles.
// Block b owns batch rows [16b, 16b+16); 8 waves each own a 16-col tile.
// LDS: W_h 64K | W_y 64K | h x2 8K | xp x2 8K  = 160 KB.
// ---------------------------------------------------------------------------
__global__ __launch_bounds__(256) void rnn_scan_fused(
    const float* __restrict__ W_h,   // 128 x 128
    const float* __restrict__ W_y,   // 128 x 128
    const float* __restrict__ b_y,   // 128
    float* __restrict__ H,           // T x N x 128 (in: xproj, out: h)
    float* __restrict__ Y)           // T x N x 128
{
    extern __shared__ float smem[];
    float* whlds = smem;                   // 16384
    float* wylds = whlds + 128 * 128;      // 16384
    float* hbuf0 = wylds + 128 * 128;      // 2048
    float* hbuf1 = hbuf0 + 16 * 128;       // 2048
    float* xpb0  = hbuf1 + 16 * 128;       // 2048
    float* xpb1  = xpb0  + 16 * 128;       // 2048
    float* xpb[2] = { xpb0, xpb1 };

    const int tid = threadIdx.x;
    const int n0  = blockIdx.x * 16;

    {   // W_h, W_y -> LDS (coalesced float4)
        const v4f* sh = (const v4f*)W_h;
        const v4f* sy = (const v4f*)W_y;
        v4f* dh = (v4f*)whlds;
        v4f* dy = (v4f*)wylds;
#pragma unroll
        for (int i = 0; i < 16; ++i) {
            dh[tid + i * 256] = sh[tid + i * 256];
            dy[tid + i * 256] = sy[tid + i * 256];
        }
    }
    {   // xp_0 -> xpb0 (coalesced)
        const v4f* g = (const v4f*)(H + (long)n0 * 128);
        v4f* d = (v4f*)xpb0;
        d[tid] = g[tid];
        d[tid + 256] = g[tid + 256];
    }
    // h_{-1} = 0
#pragma unroll
    for (int i = 0; i < 8; ++i) hbuf0[tid + i * 256] = 0.0f;
    __syncthreads();

    const int lane  = tid & 31;
    const int wave  = tid >> 5;
    const int row16 = lane & 15;
    const int khalf = (lane >> 4) << 1;
    const int hi8   = (lane >> 4) << 3;
    const int ctile = wave * 16 + row16;   // output column owned by this lane
    const float byv = b_y[ctile];

    // LDS byte offsets of the two xp buffers (flat addr low 32 bits == LDS addr)
    const unsigned int xpb_lds[2] = {
        (unsigned int)(unsigned long long)(uintptr_t)xpb0,
        (unsigned int)(unsigned long long)(uintptr_t)xpb1
    };

    float* cur = hbuf0;
    float* nxt = hbuf1;

    for (int t = 0; t <= RNN_T; ++t) {
        const bool do_pf = (t + 1 < RNN_T);

        // ---- TDM: DMA xp_{t+1} (one contiguous 8 KB tile) into LDS ----
        // Issued by wave 0 only (TDM ignores EXEC; one descriptor per block).
        if (do_pf && wave == 0) {
            const unsigned long long ga =
                (unsigned long long)(uintptr_t)(H + ((long)(t + 1) * RNN_N + n0) * 128);
            const unsigned int la = xpb_lds[(t + 1) & 1];
            // D# group 0: count=1 | lds_addr | global_addr[56:0] | type=2
            v4u g0 = { 1u, la, (unsigned int)ga,
                       (unsigned int)(ga >> 32) | (2u << 30) };
            // D# group 1: data_size=4B (2<<16); tensor_dim0=2048 (w1[31:16]);
            //             tile_dim0=2048 (w3[31:16]); tile_dim1/2=0 (1-D);
            //             tensor_dim0_stride=2048; workgroup_mask=0.
            v8i g1 = { (int)(2u << 16), (int)(2048u << 16), 0,
                       (int)(2048u << 16), 0, 2048, 0, 0 };
            v4i gz4 = { 0, 0, 0, 0 };              // groups 2/3 unused (<=2-D)
            v8i gz8 = { 0, 0, 0, 0, 0, 0, 0, 0 };  // trailing group (unused)
            __builtin_amdgcn_tensor_load_to_lds(g0, g1, gz4, gz4, gz8, 0);
        }

        // ---- dual GEMMs sharing A fragments; 2 accumulators each ----
        v8f ah0 = {}, ah1 = {}, ay0 = {}, ay1 = {};
        const float* arow  = cur   + row16 * 128 + khalf;
        const float* bhrow = whlds + ctile * 128 + khalf;
        const float* byrow = wylds + ctile * 128 + khalf;
#pragma unroll
        for (int ks = 0; ks < 32; ks += 2) {
            v2f a0  = *(const v2f*)(arow  + ks * 4);
            v2f a1  = *(const v2f*)(arow  + ks * 4 + 4);
            v2f bh0 = *(const v2f*)(bhrow + ks * 4);
            v2f bh1 = *(const v2f*)(bhrow + ks * 4 + 4);
            v2f by0 = *(const v2f*)(byrow + ks * 4);
            v2f by1 = *(const v2f*)(byrow + ks * 4 + 4);
            ah0 = WMMA4(ah0, a0, bh0);
            ah1 = WMMA4(ah1, a1, bh1);
            ay0 = WMMA4(ay0, a0, by0);
            ay1 = WMMA4(ay1, a1, by1);
        }

        if (t >= 1) {
            // y_{t-1} = h_{t-1} @ W_y^T + b_y
            float* yp = Y + (((long)(t - 1) * RNN_N) + n0 + hi8) * 128 + ctile;
#pragma unroll
            for (int r = 0; r < 8; ++r)
                yp[r * 128] = ay0[r] + ay1[r] + byv;
            // coalesced store of h_{t-1} (full tile sits in `cur`)
            const v4f* hs = (const v4f*)cur;
            v4f* hd = (v4f*)(H + ((long)(t - 1) * RNN_N + n0) * 128);
            hd[tid] = hs[tid];
            hd[tid + 256] = hs[tid + 256];
        }

        if (t < RNN_T) {
            // h_t = relu(xp_t + acc_h) -> next LDS buffer
            const float* xp = xpb[t & 1] + hi8 * 128 + ctile;
            float* lp = nxt + hi8 * 128 + ctile;
#pragma unroll
            for (int r = 0; r < 8; ++r) {
                float v = xp[r * 128] + ah0[r] + ah1[r];
                lp[r * 128] = v > 0.0f ? v : 0.0f;
            }
        }

        // DMA for xp_{t+1} must land before anyone reads it next iteration.
        __builtin_amdgcn_s_wait_tensorcnt(0);   // no-op for waves that issued none
        __syncthreads();
        float* tmp = cur; cur = nxt; nxt = tmp;
    }
}

// ---------------------------------------------------------------------------
extern "C" void kernel_launch(void* const* d_in, const int* in_sizes, int n_in,
                              void* d_out, int out_size, void* d_ws, size_t ws_size,
                              hipStream_t stream) {
    (void)in_sizes; (void)n_in; (void)out_size; (void)d_ws; (void)ws_size;

    const float* x   = (const float*)d_in[0];  // T*N*C
    const float* W_x = (const float*)d_in[1];  // D*C
    const float* b_x = (const float*)d_in[2];  // D
    const float* W_h = (const float*)d_in[3];  // D*D
    const float* W_y = (const float*)d_in[4];  // K*D
    const float* b_y = (const float*)d_in[5];  // K

    float* out   = (float*)d_out;
    float* all_y = out;                                   // T*N*K
    float* all_h = out + (long)RNN_T * RNN_N * RNN_K;     // T*N*D

    const int M = RNN_T * RNN_N;                          // 131072
    const dim3 blk(256);
    const size_t shW    = 128 * 128 * sizeof(float);                       // 64 KB
    const size_t shScan = 2 * shW + 4 * 16 * 128 * sizeof(float);          // 160 KB

    // Phase 1: xproj -> all_h region (in-place input to the scan)
    gemm_xWT_bias<<<M / 128, blk, shW, stream>>>(x, W_x, b_x, all_h);
    // Phase 2: fused scan + output projection with TDM xp prefetch
    rnn_scan_fused<<<RNN_N / 16, blk, shScan, stream>>>(W_h, W_y, b_y, all_h, all_y);
}